// GraphSAGE4WeightedMetapathMLPEdgeScorer_481036337300
// MI455X (gfx1250) — compile-verified
//
#include <hip/hip_runtime.h>
#include <math.h>

#define Nn   50000
#define Hh   128
#define Ee   500000
#define ESe  400000
#define ATS  260   // padded LDS row stride (floats) to avoid 64-bank conflicts

typedef __attribute__((ext_vector_type(2))) float v2f;
typedef __attribute__((ext_vector_type(8))) float v8f;

// ---------------------------------------------------------------- utilities
__global__ void zero_f32(float* __restrict__ p, int n) {
    int i = blockIdx.x * blockDim.x + threadIdx.x;
    if (i < n) p[i] = 0.0f;
}

__global__ void deg_count(const int* __restrict__ dst, int E, float* __restrict__ deg) {
    int e = blockIdx.x * blockDim.x + threadIdx.x;
    if (e < E) atomicAdd(&deg[dst[e]], 1.0f);
}

// one thread = (edge, 4-float chunk): gather h[src] row, atomic-add into agg[dst]
__global__ void scatter_feats(const float* __restrict__ h, const int* __restrict__ src,
                              const int* __restrict__ dst, int E, float* __restrict__ agg) {
    int idx = blockIdx.x * blockDim.x + threadIdx.x;
    int e = idx >> 5;
    if (e >= E) return;
    int c = (idx & 31) * 4;
    int s = src[e], d = dst[e];
    float4 v = *(const float4*)(h + (size_t)s * Hh + c);
    float* o = agg + (size_t)d * Hh + c;
    atomicAdd(o + 0, v.x); atomicAdd(o + 1, v.y);
    atomicAdd(o + 2, v.z); atomicAdd(o + 3, v.w);
}

// ------------------------------------------------- fused SAGE layer (WMMA f32)
// out[m, col] = act( x@Ws + b + (agg/max(deg,1))@Wn )
// grid = (N/16, 2), block = 128 (4 waves); wave w owns column tile blockIdx.y*4+w
__global__ void __launch_bounds__(128)
sage_wmma(const float* __restrict__ x, const float* __restrict__ agg,
          const float* __restrict__ deg, const float* __restrict__ Ws,
          const float* __restrict__ bias, const float* __restrict__ Wn,
          float* __restrict__ out, int doRelu)
{
    const int lane  = threadIdx.x & 31;
    const int wave  = threadIdx.x >> 5;
    const int mBase = blockIdx.x * 16;
    const int nTile = (blockIdx.y * 4 + wave) * 16;
    const bool hi   = lane >= 16;
    const int  l16  = lane & 15;
    const int  row  = mBase + l16;   // A-matrix row for this lane (all lanes)
    const int  col  = nTile + l16;   // B/C column for this lane

    const float inv = 1.0f / fmaxf(deg[row], 1.0f);
    const float bv  = bias[col];
    v8f acc = { bv, bv, bv, bv, bv, bv, bv, bv };

    const float* __restrict__ xr = x   + (size_t)row * 128;
    const float* __restrict__ ar = agg + (size_t)row * 128;
    #pragma unroll 4
    for (int k = 0; k < 128; k += 4) {
        const int k0 = k + (hi ? 2 : 0);
        v2f aS = { xr[k0],        xr[k0 + 1] };
        v2f aN = { ar[k0] * inv,  ar[k0 + 1] * inv };
        v2f bS = { Ws[(size_t)k0 * 128 + col], Ws[(size_t)(k0 + 1) * 128 + col] };
        v2f bN = { Wn[(size_t)k0 * 128 + col], Wn[(size_t)(k0 + 1) * 128 + col] };
        acc = __builtin_amdgcn_wmma_f32_16x16x4_f32(false, aS, false, bS, (short)0, acc, false, false);
        acc = __builtin_amdgcn_wmma_f32_16x16x4_f32(false, aN, false, bN, (short)0, acc, false, false);
    }
    #pragma unroll
    for (int r = 0; r < 8; ++r) {
        const int m = mBase + r + (hi ? 8 : 0);
        float v = acc[r];
        if (doRelu) v = fmaxf(v, 0.0f);
        out[(size_t)m * 128 + col] = v;
    }
}

// ------------------------------------------------- metapath attention logits
// block = 128 threads (4 waves) per node; wave i handles metapath i
__global__ void __launch_bounds__(128)
att_logits(const float* __restrict__ hstack, const float* __restrict__ attW,
           const float* __restrict__ attb, float* __restrict__ logits)
{
    const int n    = blockIdx.x;
    const int wave = threadIdx.x >> 5;
    const int lane = threadIdx.x & 31;
    const float* __restrict__ hp = hstack + ((size_t)wave * Nn + n) * Hh;
    const float* __restrict__ wp = attW + wave * Hh;
    float p = 0.0f;
    #pragma unroll
    for (int t = 0; t < 4; ++t) {
        const int k = lane * 4 + t;
        p += hp[k] * wp[k];
    }
    #pragma unroll
    for (int off = 16; off > 0; off >>= 1) p += __shfl_down(p, off, 32);
    if (lane == 0) logits[wave * Nn + n] = p + attb[wave];
}

// softmax over 4 metapaths + weighted sum -> node embeddings
__global__ void combine(const float* __restrict__ hstack, const float* __restrict__ logits,
                        float* __restrict__ emb)
{
    const int i = blockIdx.x * blockDim.x + threadIdx.x;
    if (i >= Nn * Hh) return;
    const int n = i >> 7;
    const float l0 = logits[0 * Nn + n], l1 = logits[1 * Nn + n];
    const float l2 = logits[2 * Nn + n], l3 = logits[3 * Nn + n];
    const float m  = fmaxf(fmaxf(l0, l1), fmaxf(l2, l3));
    const float w0 = __expf(l0 - m), w1 = __expf(l1 - m);
    const float w2 = __expf(l2 - m), w3 = __expf(l3 - m);
    const float inv = 1.0f / (w0 + w1 + w2 + w3);
    const size_t NH = (size_t)Nn * Hh;
    emb[i] = (w0 * hstack[i] + w1 * hstack[NH + i] +
              w2 * hstack[2 * NH + i] + w3 * hstack[3 * NH + i]) * inv;
}

// ------------------------------------------------- edge MLP scorer (WMMA f32)
// block = 256 thr (8 waves) -> 16 edges; he = relu([emb[s]||emb[d]]@W1+b1); sigmoid(he.w2+b2)
__global__ void __launch_bounds__(256)
edge_score_wmma(const float* __restrict__ emb, const int* __restrict__ src,
                const int* __restrict__ dst, const float* __restrict__ W1,
                const float* __restrict__ b1v, const float* __restrict__ w2,
                const float* __restrict__ b2s, float* __restrict__ out)
{
    __shared__ float A[16 * ATS];
    __shared__ float sc[16];
    const int t = threadIdx.x;
    const int eBase = blockIdx.x * 16;

    for (int idx = t; idx < 16 * 256; idx += 256) {
        const int e = idx >> 8;
        const int k = idx & 255;
        const int node = (k < 128) ? src[eBase + e] : dst[eBase + e];
        A[e * ATS + k] = emb[(size_t)node * 128 + (k & 127)];
    }
    if (t < 16) sc[t] = 0.0f;
    __syncthreads();

    const int  lane = t & 31;
    const int  wave = t >> 5;
    const bool hi   = lane >= 16;
    const int  l16  = lane & 15;
    const int  col  = wave * 16 + l16;

    const float bv = b1v[col];
    v8f acc = { bv, bv, bv, bv, bv, bv, bv, bv };
    #pragma unroll 4
    for (int k = 0; k < 256; k += 4) {
        const int k0 = k + (hi ? 2 : 0);
        v2f a = { A[l16 * ATS + k0], A[l16 * ATS + k0 + 1] };
        v2f b = { W1[(size_t)k0 * 128 + col], W1[(size_t)(k0 + 1) * 128 + col] };
        acc = __builtin_amdgcn_wmma_f32_16x16x4_f32(false, a, false, b, (short)0, acc, false, false);
    }
    const float wc = w2[col];
    #pragma unroll
    for (int r = 0; r < 8; ++r) {
        const int m = r + (hi ? 8 : 0);
        atomicAdd(&sc[m], fmaxf(acc[r], 0.0f) * wc);
    }
    __syncthreads();
    if (t < 16) {
        const float s = sc[t] + b2s[0];
        out[eBase + t] = 1.0f / (1.0f + __expf(-s));
    }
}

// ---------------------------------------------------------------- launcher
extern "C" void kernel_launch(void* const* d_in, const int* in_sizes, int n_in,
                              void* d_out, int out_size, void* d_ws, size_t ws_size,
                              hipStream_t stream) {
    const float* nodef = (const float*)d_in[0];
    const float* Ws1   = (const float*)d_in[1];
    const float* b1    = (const float*)d_in[2];
    const float* Wn1   = (const float*)d_in[3];
    const float* Ws2   = (const float*)d_in[4];
    const float* b2    = (const float*)d_in[5];
    const float* Wn2   = (const float*)d_in[6];
    const float* attW  = (const float*)d_in[7];
    const float* attb  = (const float*)d_in[8];
    const float* l1W   = (const float*)d_in[9];
    const float* l1b   = (const float*)d_in[10];
    const float* l2W   = (const float*)d_in[11];
    const float* l2b   = (const float*)d_in[12];
    const int* srcs[4] = { (const int*)d_in[13], (const int*)d_in[15],
                           (const int*)d_in[17], (const int*)d_in[19] };
    const int* dsts[4] = { (const int*)d_in[14], (const int*)d_in[16],
                           (const int*)d_in[18], (const int*)d_in[20] };
    const int* ssim = (const int*)d_in[21];
    const int* dsim = (const int*)d_in[22];
    float* out = (float*)d_out;

    float* ws = (float*)d_ws;
    const size_t NH = (size_t)Nn * Hh;
    float* hstack = ws;             // 4*NH
    float* h1     = ws + 4 * NH;    // NH
    float* agg    = ws + 5 * NH;    // NH
    float* emb    = ws + 6 * NH;    // NH
    float* deg    = ws + 7 * NH;    // Nn
    float* logits = deg + Nn;       // 4*Nn

    const dim3 sageGrid(Nn / 16, 2);
    const int scatterBlocks = (Ee * 32 + 255) / 256;

    for (int i = 0; i < 4; ++i) {
        zero_f32<<<(int)((NH + 255) / 256), 256, 0, stream>>>(agg, (int)NH);
        zero_f32<<<(Nn + 255) / 256, 256, 0, stream>>>(deg, Nn);
        deg_count<<<(Ee + 255) / 256, 256, 0, stream>>>(dsts[i], Ee, deg);
        scatter_feats<<<scatterBlocks, 256, 0, stream>>>(nodef, srcs[i], dsts[i], Ee, agg);
        sage_wmma<<<sageGrid, 128, 0, stream>>>(nodef, agg, deg,
            Ws1 + (size_t)i * 128 * 128, b1 + i * 128, Wn1 + (size_t)i * 128 * 128, h1, 1);
        zero_f32<<<(int)((NH + 255) / 256), 256, 0, stream>>>(agg, (int)NH);
        scatter_feats<<<scatterBlocks, 256, 0, stream>>>(h1, srcs[i], dsts[i], Ee, agg);
        sage_wmma<<<sageGrid, 128, 0, stream>>>(h1, agg, deg,
            Ws2 + (size_t)i * 128 * 128, b2 + i * 128, Wn2 + (size_t)i * 128 * 128,
            hstack + i * NH, 0);
    }
    att_logits<<<Nn, 128, 0, stream>>>(hstack, attW, attb, logits);
    combine<<<(int)((NH + 255) / 256), 256, 0, stream>>>(hstack, logits, emb);
    edge_score_wmma<<<ESe / 16, 256, 0, stream>>>(emb, ssim, dsim, l1W, l1b, l2W, l2b, out);
}